// InstanceBank_335007449262
// MI455X (gfx1250) — compile-verified
//
#include <hip/hip_runtime.h>
#include <stdint.h>

// Sparse4D InstanceBank dims
#define B_     128
#define NA_    900
#define NT_    600
#define NC_    10
#define C_     256
#define D_     11
#define NSEL_  300     // NA - NT
#define DECAY_ 0.6f

// ---------------------------------------------------------------------------
// CDNA5 (gfx1250) async global<->LDS DMA helpers.
// Per-lane: each instruction moves 16B between the LDS offset in lds_off and the global address in gaddr;
// a full wave32 moves 512B per instruction. Tracked with ASYNCcnt.
// Generic pointers to __shared__ carry the LDS byte offset in their low 32
// bits, so truncation yields VDST/VSRC.
// ---------------------------------------------------------------------------
__device__ __forceinline__ void async_load_b128(uint32_t lds_off, uint64_t gaddr) {
  asm volatile("global_load_async_to_lds_b128 %0, %1, off"
               :: "v"(lds_off), "v"(gaddr) : "memory");
}
__device__ __forceinline__ void async_store_b128(uint64_t gaddr, uint32_t lds_off) {
  asm volatile("global_store_async_from_lds_b128 %0, %1, off"
               :: "v"(gaddr), "v"(lds_off) : "memory");
}
__device__ __forceinline__ void wait_asynccnt0() {
#if __has_builtin(__builtin_amdgcn_s_wait_asynccnt)
  __builtin_amdgcn_s_wait_asynccnt(0);
#else
  asm volatile("s_wait_asynccnt 0x0" ::: "memory");
#endif
}

// ---------------------------------------------------------------------------
// K1: per-batch class-max, sigmoid+decay, and both stable top-k rankings.
// One block per batch element. Exact jax.lax.top_k semantics:
// descending order, ties broken by lower index (stable rank is a bijection,
// so every output slot is written exactly once -> deterministic).
// ---------------------------------------------------------------------------
__global__ void topk_rank_kernel(const float* __restrict__ confidence,   // [B,NA,NC]
                                 const float* __restrict__ cached_conf,  // [B,NT]
                                 int* __restrict__ sel_idx,              // [B,NSEL]
                                 int* __restrict__ cidx,                 // [B,NT]
                                 float* __restrict__ new_conf) {         // [B,NT]
  __shared__ float sconf[NA_];   // max-class logit
  __shared__ float scach[NA_];   // sigmoid (+decay-max on temporal slots)
  const int b = blockIdx.x;

  for (int i = threadIdx.x; i < NA_; i += blockDim.x) {
    const float* p = confidence + ((size_t)b * NA_ + i) * NC_;
    float m = p[0];
#pragma unroll
    for (int k = 1; k < NC_; ++k) m = fmaxf(m, p[k]);
    sconf[i] = m;
    float s = 1.0f / (1.0f + __expf(-m));
    if (i < NT_) s = fmaxf(cached_conf[(size_t)b * NT_ + i] * DECAY_, s);
    scach[i] = s;
  }
  __syncthreads();

  for (int i = threadIdx.x; i < NA_; i += blockDim.x) {
    const float v = sconf[i];
    const float w = scach[i];
    int r1 = 0, r2 = 0;
    for (int j = 0; j < NA_; ++j) {           // broadcast LDS reads
      const float a = sconf[j];
      const float c = scach[j];
      r1 += (int)((a > v) || ((a == v) && (j < i)));
      r2 += (int)((c > w) || ((c == w) && (j < i)));
    }
    if (r1 < NSEL_) sel_idx[(size_t)b * NSEL_ + r1] = i;
    if (r2 < NT_) {
      cidx[(size_t)b * NT_ + r2] = i;
      new_conf[(size_t)b * NT_ + r2] = w;
    }
  }
}

// ---------------------------------------------------------------------------
// K2: out_feat [B,NA,C]. One wave per 1KB row; async DMA through LDS.
// Row n<NT  : cached_feature (if mask) ; n>=NT : gathered top-NSEL instance
// row (if mask) ; else passthrough instance_feature.
// ---------------------------------------------------------------------------
__global__ void feat_merge_kernel(const float* __restrict__ instance_feature,
                                  const float* __restrict__ cached_feature,
                                  const int* __restrict__ sel_idx,
                                  const unsigned char* __restrict__ mask,
                                  float* __restrict__ out_feat) {
  __shared__ __align__(16) float buf[8][C_];      // 1KB per wave, 8 waves/block
  const int wave = threadIdx.x >> 5;
  const int lane = threadIdx.x & 31;
  const int row  = blockIdx.x * 8 + wave;         // [0, B*NA)
  if (row >= B_ * NA_) return;
  const int b = row / NA_;
  const int n = row - b * NA_;

  const float* src;
  if (mask[b]) {
    if (n < NT_) {
      src = cached_feature + ((size_t)b * NT_ + n) * C_;
    } else {
      const int s = sel_idx[(size_t)b * NSEL_ + (n - NT_)];
      src = instance_feature + ((size_t)b * NA_ + s) * C_;
    }
  } else {
    src = instance_feature + (size_t)row * C_;
  }
  float* dst = out_feat + (size_t)row * C_;

  const uint32_t l0 = (uint32_t)(uintptr_t)(&buf[wave][0]) + (uint32_t)lane * 16u;
  const uint64_t s0 = (uint64_t)(uintptr_t)src + (uint64_t)lane * 16u;
  const uint64_t d0 = (uint64_t)(uintptr_t)dst + (uint64_t)lane * 16u;

  async_load_b128(l0,        s0);
  async_load_b128(l0 + 512u, s0 + 512u);
  wait_asynccnt0();                   // loads landed in LDS
  async_store_b128(d0,        l0);
  async_store_b128(d0 + 512u, l0 + 512u);
  wait_asynccnt0();                   // drain before LDS dealloc at endpgm
}

// ---------------------------------------------------------------------------
// K3: out_anc [B,NA,D], D=11 -> elementwise.
// ---------------------------------------------------------------------------
__global__ void anc_merge_kernel(const float* __restrict__ anchor,
                                 const float* __restrict__ cached_anchor,
                                 const int* __restrict__ sel_idx,
                                 const unsigned char* __restrict__ mask,
                                 float* __restrict__ out_anc) {
  const int tid = blockIdx.x * blockDim.x + threadIdx.x;
  const int total = B_ * NA_ * D_;
  if (tid >= total) return;
  const int d   = tid % D_;
  const int row = tid / D_;
  const int b   = row / NA_;
  const int n   = row - b * NA_;
  float v;
  if (mask[b]) {
    if (n < NT_) {
      v = cached_anchor[((size_t)b * NT_ + n) * D_ + d];
    } else {
      const int s = sel_idx[(size_t)b * NSEL_ + (n - NT_)];
      v = anchor[((size_t)b * NA_ + s) * D_ + d];
    }
  } else {
    v = anchor[(size_t)row * D_ + d];
  }
  out_anc[tid] = v;
}

// ---------------------------------------------------------------------------
// K4: new_cached_feat [B,NT,C] = gather rows of out_feat at cidx.
// out_feat (118MB) < 192MB L2 -> re-reads should be L2 hits. Async DMA rows.
// ---------------------------------------------------------------------------
__global__ void cache_feat_kernel(const float* __restrict__ out_feat,
                                  const int* __restrict__ cidx,
                                  float* __restrict__ new_cached_feat) {
  __shared__ __align__(16) float buf[8][C_];
  const int wave = threadIdx.x >> 5;
  const int lane = threadIdx.x & 31;
  const int row  = blockIdx.x * 8 + wave;         // [0, B*NT)
  if (row >= B_ * NT_) return;
  const int b = row / NT_;
  const int t = row - b * NT_;
  const int s = cidx[(size_t)b * NT_ + t];

  const float* src = out_feat + ((size_t)b * NA_ + s) * C_;
  float*       dst = new_cached_feat + (size_t)row * C_;

  const uint32_t l0 = (uint32_t)(uintptr_t)(&buf[wave][0]) + (uint32_t)lane * 16u;
  const uint64_t s0 = (uint64_t)(uintptr_t)src + (uint64_t)lane * 16u;
  const uint64_t d0 = (uint64_t)(uintptr_t)dst + (uint64_t)lane * 16u;

  async_load_b128(l0,        s0);
  async_load_b128(l0 + 512u, s0 + 512u);
  wait_asynccnt0();
  async_store_b128(d0,        l0);
  async_store_b128(d0 + 512u, l0 + 512u);
  wait_asynccnt0();
}

// ---------------------------------------------------------------------------
// K5: new_cached_anc [B,NT,D] = gather rows of out_anc at cidx (elementwise).
// ---------------------------------------------------------------------------
__global__ void cache_anc_kernel(const float* __restrict__ out_anc,
                                 const int* __restrict__ cidx,
                                 float* __restrict__ new_cached_anc) {
  const int tid = blockIdx.x * blockDim.x + threadIdx.x;
  const int total = B_ * NT_ * D_;
  if (tid >= total) return;
  const int d   = tid % D_;
  const int row = tid / D_;
  const int b   = row / NT_;
  const int t   = row - b * NT_;
  const int s   = cidx[(size_t)b * NT_ + t];
  new_cached_anc[tid] = out_anc[((size_t)b * NA_ + s) * D_ + d];
}

// ---------------------------------------------------------------------------
extern "C" void kernel_launch(void* const* d_in, const int* in_sizes, int n_in,
                              void* d_out, int out_size, void* d_ws, size_t ws_size,
                              hipStream_t stream) {
  const float*         instance_feature = (const float*)d_in[0];  // [B,NA,C]
  const float*         anchor           = (const float*)d_in[1];  // [B,NA,D]
  const float*         confidence       = (const float*)d_in[2];  // [B,NA,NC]
  const float*         cached_feature   = (const float*)d_in[3];  // [B,NT,C]
  const float*         cached_anchor    = (const float*)d_in[4];  // [B,NT,D]
  const float*         cached_conf      = (const float*)d_in[5];  // [B,NT]
  const unsigned char* mask             = (const unsigned char*)d_in[6];  // [B] bool

  // outputs, concatenated flat in reference return order
  float* out_feat  = (float*)d_out;                               // B*NA*C
  float* out_anc   = out_feat  + (size_t)B_ * NA_ * C_;           // B*NA*D
  float* new_conf  = out_anc   + (size_t)B_ * NA_ * D_;           // B*NT
  float* new_cfeat = new_conf  + (size_t)B_ * NT_;                // B*NT*C
  float* new_canc  = new_cfeat + (size_t)B_ * NT_ * C_;           // B*NT*D

  // workspace: selection indices
  int* sel_idx = (int*)d_ws;                                      // B*NSEL
  int* cidx    = sel_idx + (size_t)B_ * NSEL_;                    // B*NT

  // K1: rankings (writes sel_idx, cidx, new_conf)
  hipLaunchKernelGGL(topk_rank_kernel, dim3(B_), dim3(256), 0, stream,
                     confidence, cached_conf, sel_idx, cidx, new_conf);

  // K2: out_feat merge (1 wave / 1KB row, async LDS DMA)
  hipLaunchKernelGGL(feat_merge_kernel, dim3((B_ * NA_) / 8), dim3(256), 0, stream,
                     instance_feature, cached_feature, sel_idx, mask, out_feat);

  // K3: out_anc merge
  {
    const int total = B_ * NA_ * D_;
    hipLaunchKernelGGL(anc_merge_kernel, dim3((total + 255) / 256), dim3(256), 0, stream,
                       anchor, cached_anchor, sel_idx, mask, out_anc);
  }

  // K4: cached-feature gather from out_feat (L2-resident)
  hipLaunchKernelGGL(cache_feat_kernel, dim3((B_ * NT_) / 8), dim3(256), 0, stream,
                     out_feat, cidx, new_cfeat);

  // K5: cached-anchor gather from out_anc
  {
    const int total = B_ * NT_ * D_;
    hipLaunchKernelGGL(cache_anc_kernel, dim3((total + 255) / 256), dim3(256), 0, stream,
                       out_anc, cidx, new_canc);
  }
}